// TripletSemihardLoss_1580547970270
// MI455X (gfx1250) — compile-verified
//
#include <hip/hip_runtime.h>
#include <hip/hip_bf16.h>
#include <math.h>

#define N    256
#define DIM  128
#define EPSD 1e-4f
#define MARGIN 1.0f

typedef __attribute__((ext_vector_type(2))) float v2f;
typedef __attribute__((ext_vector_type(8))) float v8f;

// ---------------------------------------------------------------------------
// Kernel 1: row norms  norms[i] = sum_k E[i][k]^2   (single block, 256 thr)
// ---------------------------------------------------------------------------
__global__ __launch_bounds__(N) void norms_kernel(const float* __restrict__ E,
                                                  float* __restrict__ norms) {
    const int i = threadIdx.x;
    const float* row = E + i * DIM;
    float s = 0.0f;
#pragma unroll 4
    for (int k = 0; k < DIM; ++k) {
        float v = row[k];
        s = fmaf(v, v, s);
    }
    norms[i] = s;
}

// ---------------------------------------------------------------------------
// Kernel 2: pairwise distance matrix via fp32 WMMA.
// One wave computes one 16x16 tile of prod = E * E^T, then applies the
// distance epilogue d = sqrt(max(n_i + n_j - 2*prod, EPS)) in-register.
//
// V_WMMA_F32_16X16X4_F32 operand layouts (wave32):
//   A (16x4):  lanes 0-15 -> M=lane, K = k0+{0,1}; lanes 16-31 -> K = k0+{2,3}
//   B (4x16):  lanes 0-15 -> N=lane, K = k0+{0,1}; lanes 16-31 -> K = k0+{2,3}
//              (B = E^T, so B[k][n] = E[colbase+n][k] -> same addressing as A)
//   C (16x16): VGPR r -> row = r + 8*(lane>=16), col = lane&15
// ---------------------------------------------------------------------------
__global__ __launch_bounds__(256) void dist_wmma_kernel(const float* __restrict__ E,
                                                        const float* __restrict__ norms,
                                                        float* __restrict__ D) {
    const int lane = threadIdx.x & 31;
    const int wave = threadIdx.x >> 5;
    const int tile = blockIdx.x * 8 + wave;      // 0..255 : 16x16 grid of tiles
    const int tm = (tile >> 4) * 16;             // output row-tile base
    const int tn = (tile & 15) * 16;             // output col-tile base

    const int half = lane >> 4;                  // 0: K pair {0,1}, 1: {2,3}
    const int l15  = lane & 15;

    const float* arow = E + (tm + l15) * DIM + 2 * half;  // A rows
    const float* brow = E + (tn + l15) * DIM + 2 * half;  // B "rows" (cols of D)

    v8f c = {};
#pragma unroll 8
    for (int k0 = 0; k0 < DIM; k0 += 4) {
        v2f a = *(const v2f*)(arow + k0);
        v2f b = *(const v2f*)(brow + k0);
        // 8 args: (neg_a, A, neg_b, B, c_mod, C, reuse_a, reuse_b)
        c = __builtin_amdgcn_wmma_f32_16x16x4_f32(false, a, false, b,
                                                  (short)0, c, false, false);
    }

    // Epilogue: convert Gram tile to distances and store.
    const int col = tn + l15;
    const float ncol = norms[col];
#pragma unroll
    for (int r = 0; r < 8; ++r) {
        const int row = tm + r + 8 * half;
        float res = norms[row] + ncol - 2.0f * c[r];
        res = fmaxf(res, EPSD);                  // clip(.,0) then clip(.,EPS) == max(.,EPS)
        D[row * N + col] = sqrtf(res);
    }
}

// ---------------------------------------------------------------------------
// Kernel 3: per-row semi-hard selection + loss partials.
// Block i handles anchor row i; d[i][*] and the negative mask live in LDS.
// Negatives = {j : label[j] != label[i]} U {i}   (reference keeps j==i, with
// d[i][i] = sqrt(EPS)).
// partial[i]       = sum over positives k of relu(semi_hard + MARGIN)
// partial[N + i]   = number of positives in row i
// ---------------------------------------------------------------------------
__global__ __launch_bounds__(N) void row_loss_kernel(const float* __restrict__ D,
                                                     const int* __restrict__ labels,
                                                     float* __restrict__ partial) {
    __shared__ float sd[N];
    __shared__ unsigned char sneg[N];
    __shared__ float red[N];

    const int i = blockIdx.x;
    const int t = threadIdx.x;
    const int li = labels[i];
    const int lt = labels[t];

    const float dij = D[i * N + t];
    const bool isNeg = (lt != li) || (t == i);
    sd[t]   = dij;
    sneg[t] = isNeg ? 1 : 0;
    red[t]  = isNeg ? dij : -__builtin_inff();
    __syncthreads();

    // max negative distance for row i (fallback "easiest negative")
    for (int s = N / 2; s > 0; s >>= 1) {
        if (t < s) red[t] = fmaxf(red[t], red[t + s]);
        __syncthreads();
    }
    const float maxNeg = red[0];
    __syncthreads();

    // thread t plays the role of positive index k
    const bool isPos = (lt == li) && (t != i);
    float loss = 0.0f;
    if (isPos) {
        const float dik = sd[t];
        float mn = __builtin_inff();
        for (int j = 0; j < N; ++j) {
            const float dj = sd[j];
            // semi-hard: negative j with d[i,j] strictly greater than d[i,k]
            if (sneg[j] && (dj > dik)) mn = fminf(mn, dj);
        }
        const float semi = (mn < __builtin_inff()) ? (dik - mn) : (dik - maxNeg);
        loss = fmaxf(semi + MARGIN, 0.0f);
    }

    // reduce loss
    red[t] = loss;
    __syncthreads();
    for (int s = N / 2; s > 0; s >>= 1) {
        if (t < s) red[t] += red[t + s];
        __syncthreads();
    }
    if (t == 0) partial[i] = red[0];
    __syncthreads();

    // reduce positive count
    red[t] = isPos ? 1.0f : 0.0f;
    __syncthreads();
    for (int s = N / 2; s > 0; s >>= 1) {
        if (t < s) red[t] += red[t + s];
        __syncthreads();
    }
    if (t == 0) partial[N + i] = red[0];
}

// ---------------------------------------------------------------------------
// Kernel 4: final reduction  out = sum(loss partials) / sum(pos partials)
// ---------------------------------------------------------------------------
__global__ __launch_bounds__(N) void finalize_kernel(const float* __restrict__ partial,
                                                     float* __restrict__ out) {
    __shared__ float sl[N];
    __shared__ float sp[N];
    const int t = threadIdx.x;
    sl[t] = partial[t];
    sp[t] = partial[N + t];
    __syncthreads();
    for (int s = N / 2; s > 0; s >>= 1) {
        if (t < s) { sl[t] += sl[t + s]; sp[t] += sp[t + s]; }
        __syncthreads();
    }
    if (t == 0) out[0] = sl[0] / sp[0];
}

// ---------------------------------------------------------------------------
// Launch
// ---------------------------------------------------------------------------
extern "C" void kernel_launch(void* const* d_in, const int* in_sizes, int n_in,
                              void* d_out, int out_size, void* d_ws, size_t ws_size,
                              hipStream_t stream) {
    const float* E      = (const float*)d_in[0];  // [256,128] fp32
    const int*   labels = (const int*)d_in[1];    // [256] int
    float*       out    = (float*)d_out;          // scalar

    // workspace layout (floats): norms[256] | D[256*256] | partial[512]
    float* norms   = (float*)d_ws;
    float* D       = norms + N;
    float* partial = D + N * N;

    norms_kernel<<<1, N, 0, stream>>>(E, norms);
    dist_wmma_kernel<<<32, 256, 0, stream>>>(E, norms, D);   // 256 waves = 256 tiles
    row_loss_kernel<<<N, N, 0, stream>>>(D, labels, partial);
    finalize_kernel<<<1, N, 0, stream>>>(partial, out);
}